// FusionModule_36601711296528
// MI455X (gfx1250) — compile-verified
//
#include <hip/hip_runtime.h>
#include <math.h>

// ---------------------------------------------------------------------------
// FusionModule for MI455X (gfx1250): all GEMM-shaped work (im2col convs,
// deformable einsum, 1x1 encoder, fusion conv) runs through a generic
// V_WMMA_F32_16X16X4_F32 tile kernel (fp32 matrix path -> keeps fp32
// reference precision).  Each wave owns a 32x64 D tile (8 accumulators):
// per K-step two A fragments are shared by 8 WMMAs; K is ordered (ky,kx,c)
// so the channel inner loop is division-free with constant-stride,
// branchless (clamped+selected) loads plus global_prefetch on the streamed
// walks.  Irregular ops (pool, correlation, bilinear sampling, tiny FCs)
// are VALU kernels; they are bandwidth-trivial.
// ---------------------------------------------------------------------------

typedef float v2f __attribute__((ext_vector_type(2)));
typedef float v8f __attribute__((ext_vector_type(8)));

// Problem constants (match reference setup_inputs)
#define BB   4
#define CC   256
#define HH   64
#define WW   64
#define HW   (HH*WW)          // 4096
#define DIMS 128
#define KK3  3
#define CORR 49               // (2*3+1)^2 displacement taps
#define IFWC (CORR + 2*CC)    // 561
#define IFOC (CORR + CC)      // 305
#define WROW (DIMS*DIMS*KK3*KK3)  // 147456
#define DKK  (DIMS*KK3*KK3)       // 1152

// ---------------------------------------------------------------------------
// 2x2 average pool: dst[b,c,h,w] = mean of src[b,sel,c,2h:2h+2,2w:2w+2]
// ---------------------------------------------------------------------------
__global__ void pool_kernel(const float* __restrict__ src,
                            float* __restrict__ dst, int sel) {
  int idx = blockIdx.x * blockDim.x + threadIdx.x;
  if (idx >= BB*CC*HW) return;
  int w = idx & 63;
  int h = (idx >> 6) & 63;
  int c = (idx >> 12) & 255;
  int b = idx >> 20;
  const float* p = src + (((size_t)(b*2 + sel)*CC + c)*(2*HH) + 2*h)*(2*WW) + 2*w;
  dst[idx] = 0.25f * (p[0] + p[1] + p[2*WW] + p[2*WW + 1]);
}

// ---------------------------------------------------------------------------
// correlation: out[b,t,h,w] = (1/C) sum_c A[b,c,h,w] * B[b,c,h+dy,w+dx]
// t = (i+3)*7 + (j+3), dy = 2i, dx = 2j; zero outside.
// ---------------------------------------------------------------------------
__global__ void corr_kernel(const float* __restrict__ A,
                            const float* __restrict__ Bt,
                            float* __restrict__ out) {
  int idx = blockIdx.x * blockDim.x + threadIdx.x;
  if (idx >= BB*CORR*HW) return;
  int p = idx & (HW - 1);
  int t = (idx >> 12) % CORR;
  int b = idx / (CORR*HW);
  int w = p & 63, h = p >> 6;
  int dy = 2 * ((t / 7) - 3);
  int dx = 2 * ((t % 7) - 3);
  int hh = h + dy, ww = w + dx;
  float s = 0.f;
  if (hh >= 0 && hh < HH && ww >= 0 && ww < WW) {
    const float* pa = A  + (size_t)b*CC*HW + p;
    const float* pb = Bt + (size_t)b*CC*HW + hh*WW + ww;
    for (int c = 0; c < CC; ++c) s += pa[(size_t)c*HW] * pb[(size_t)c*HW];
  }
  out[idx] = s * (1.0f / (float)CC);
}

// ---------------------------------------------------------------------------
// concat -> ifw (B,561,H,W): [corr(49), inputs[2b+first](256), inputs[2b+1-first](256)]
// ---------------------------------------------------------------------------
__global__ void concat_kernel(const float* __restrict__ corr,
                              const float* __restrict__ inputs,
                              int firstSel, float* __restrict__ ifw) {
  int idx = blockIdx.x * blockDim.x + threadIdx.x;
  if (idx >= BB*IFWC*HW) return;
  int p = idx & (HW - 1);
  int c = (idx >> 12) % IFWC;
  int b = idx / (IFWC*HW);
  float val;
  if (c < CORR)
    val = corr[((size_t)b*CORR + c)*HW + p];
  else if (c < IFOC)
    val = inputs[((size_t)(2*b + firstSel)*CC + (c - CORR))*HW + p];
  else
    val = inputs[((size_t)(2*b + 1 - firstSel)*CC + (c - IFOC))*HW + p];
  ifw[idx] = val;
}

// ---------------------------------------------------------------------------
// Implicit-im2col conv as WMMA f32 GEMM, K ordered (ky,kx,c):
//   D[o, n] = sum_{ky,kx,c} wt[o,c,ky,kx] * in[c, oy*s-p+ky, ox*s-p+kx]
// Wave = 32(M) x 64(N) D tile, 8 accumulators; B fragments are reused by
// both M-subtiles, so each K-step is 8 WMMAs for 12 loads (~10.7 FLOP/B).
// N (=outH*outW) must be a multiple of 64 (true for every call site).
// The c-inner loop is division-free, constant-stride, and branchless:
// spatially-invalid taps load from clamped addresses and are zeroed with
// selects; M-overhang rows only clamp the A address (their D rows are
// never stored); the inC%4 tail is a single peeled masked step.  EXEC
// stays all-ones around every WMMA.  Fragment layouts per CDNA5 ISA:
//   A 16x4: lane<16 -> {K=k0,k0+1}, lane>=16 -> {K=k0+2,k0+3}, M=m0+lane%16
//   B 4x16: mirrored, N = base+lane%16
//   D: vgpr r -> row m0 + r + 8*(lane>=16)
// ---------------------------------------------------------------------------
__global__ __launch_bounds__(32) void conv_gemm_wmma(
    const float* __restrict__ in,  int inBatchStride,
    const float* __restrict__ wt,  int wtBatchStride,
    const float* __restrict__ bias, int biasBatchStride,
    float* __restrict__ out, int outBatchStride,
    int inC, int inH, int inW,
    int outC, int outH, int outW,
    int KH, int KW, int stride, int pad, int relu)
{
  const int lane = threadIdx.x;
  const int half = lane >> 4;
  const int lid  = lane & 15;
  const int m0 = blockIdx.x * 32;
  const int n0 = blockIdx.y * 64;
  const int N  = outH * outW;
  const int inHW = inH * inW;
  const int KKw  = KH * KW;

  const float* inB = in + (size_t)blockIdx.z * (size_t)inBatchStride;
  const float* wtB = wt + (size_t)blockIdx.z * (size_t)wtBatchStride;

  // A rows: rows >= outC are never stored, so only clamp the address.
  const int mr0 = m0 + lid;
  const int mr1 = m0 + 16 + lid;
  const int mc0 = mr0 < outC ? mr0 : outC - 1;
  const int mc1 = mr1 < outC ? mr1 : outC - 1;

  // per-group output coordinates (4 N-subtiles of 16)
  int oy[4], ox[4];
#pragma unroll
  for (int g = 0; g < 4; ++g) {
    int n = n0 + 16*g + lid;
    oy[g] = n / outW;
    ox[g] = n - oy[g]*outW;
  }

  v8f acc00 = {}, acc01 = {}, acc02 = {}, acc03 = {};
  v8f acc10 = {}, acc11 = {}, acc12 = {}, acc13 = {};
  const int inCmain = inC & ~3;

  for (int ky = 0; ky < KH; ++ky) {
    for (int kx = 0; kx < KW; ++kx) {
      const int rpos = ky*KW + kx;
      // spatial validity + clamped in-image offset per group (per lane)
      int  soff[4];
      bool sval[4];
#pragma unroll
      for (int g = 0; g < 4; ++g) {
        int iy = oy[g]*stride - pad + ky;
        int ix = ox[g]*stride - pad + kx;
        sval[g] = (iy >= 0) & (iy < inH) & (ix >= 0) & (ix < inW);
        int iyc = iy < 0 ? 0 : (iy >= inH ? inH-1 : iy);
        int ixc = ix < 0 ? 0 : (ix >= inW ? inW-1 : ix);
        soff[g] = iyc*inW + ixc;
      }

      // A walks: wt[(mc*inC + c)*KKw + rpos], c = c0 + 2*half (+1 for .y)
      const float* pA0 = wtB + ((size_t)mc0*inC + 2*half)*KKw + rpos;
      const float* pA1 = wtB + ((size_t)mc1*inC + 2*half)*KKw + rpos;
      // B walk: in[(c)*inHW + soff], c = c0 + 2*half (+1 for .y)
      const float* pB  = inB + (size_t)(2*half)*inHW;

      int c0 = 0;
#pragma unroll 2
      for (; c0 < inCmain; c0 += 4) {
        // stream-ahead prefetch (speculative; safe past end per ISA 10.5)
        __builtin_prefetch((const void*)(pB + (size_t)32*inHW), 0, 0);
        __builtin_prefetch((const void*)(pA0 + (size_t)32*KKw), 0, 0);
        v2f a0, a1;
        a0.x = pA0[0];
        a0.y = pA0[KKw];
        a1.x = pA1[0];
        a1.y = pA1[KKw];
        float t0x = pB[soff[0]], t0y = pB[inHW + soff[0]];
        float t1x = pB[soff[1]], t1y = pB[inHW + soff[1]];
        float t2x = pB[soff[2]], t2y = pB[inHW + soff[2]];
        float t3x = pB[soff[3]], t3y = pB[inHW + soff[3]];
        v2f b0, b1, b2, b3;
        b0.x = sval[0] ? t0x : 0.f;  b0.y = sval[0] ? t0y : 0.f;
        b1.x = sval[1] ? t1x : 0.f;  b1.y = sval[1] ? t1y : 0.f;
        b2.x = sval[2] ? t2x : 0.f;  b2.y = sval[2] ? t2y : 0.f;
        b3.x = sval[3] ? t3x : 0.f;  b3.y = sval[3] ? t3y : 0.f;
        acc00 = __builtin_amdgcn_wmma_f32_16x16x4_f32(false, a0, false, b0, (short)0, acc00, false, false);
        acc01 = __builtin_amdgcn_wmma_f32_16x16x4_f32(false, a0, false, b1, (short)0, acc01, false, false);
        acc02 = __builtin_amdgcn_wmma_f32_16x16x4_f32(false, a0, false, b2, (short)0, acc02, false, false);
        acc03 = __builtin_amdgcn_wmma_f32_16x16x4_f32(false, a0, false, b3, (short)0, acc03, false, false);
        acc10 = __builtin_amdgcn_wmma_f32_16x16x4_f32(false, a1, false, b0, (short)0, acc10, false, false);
        acc11 = __builtin_amdgcn_wmma_f32_16x16x4_f32(false, a1, false, b1, (short)0, acc11, false, false);
        acc12 = __builtin_amdgcn_wmma_f32_16x16x4_f32(false, a1, false, b2, (short)0, acc12, false, false);
        acc13 = __builtin_amdgcn_wmma_f32_16x16x4_f32(false, a1, false, b3, (short)0, acc13, false, false);
        pA0 += (size_t)4*KKw;
        pA1 += (size_t)4*KKw;
        pB  += (size_t)4*inHW;
      }

      if (c0 < inC) {  // peeled tail: inC % 4 in {1,2,3}
        const int ca  = c0 + 2*half;
        const bool ok0 = ca     < inC;
        const bool ok1 = ca + 1 < inC;
        const int cc0 = ok0 ? ca     : inC - 1;
        const int cc1 = ok1 ? ca + 1 : inC - 1;
        v2f a0, a1;
        a0.x = wtB[((size_t)mc0*inC + cc0)*KKw + rpos];  // B is masked, A clamp suffices
        a0.y = wtB[((size_t)mc0*inC + cc1)*KKw + rpos];
        a1.x = wtB[((size_t)mc1*inC + cc0)*KKw + rpos];
        a1.y = wtB[((size_t)mc1*inC + cc1)*KKw + rpos];
        const float* q0 = inB + (size_t)cc0*inHW;
        const float* q1 = inB + (size_t)cc1*inHW;
        v2f b0, b1, b2, b3;
        b0.x = (ok0 & sval[0]) ? q0[soff[0]] : 0.f;  b0.y = (ok1 & sval[0]) ? q1[soff[0]] : 0.f;
        b1.x = (ok0 & sval[1]) ? q0[soff[1]] : 0.f;  b1.y = (ok1 & sval[1]) ? q1[soff[1]] : 0.f;
        b2.x = (ok0 & sval[2]) ? q0[soff[2]] : 0.f;  b2.y = (ok1 & sval[2]) ? q1[soff[2]] : 0.f;
        b3.x = (ok0 & sval[3]) ? q0[soff[3]] : 0.f;  b3.y = (ok1 & sval[3]) ? q1[soff[3]] : 0.f;
        acc00 = __builtin_amdgcn_wmma_f32_16x16x4_f32(false, a0, false, b0, (short)0, acc00, false, false);
        acc01 = __builtin_amdgcn_wmma_f32_16x16x4_f32(false, a0, false, b1, (short)0, acc01, false, false);
        acc02 = __builtin_amdgcn_wmma_f32_16x16x4_f32(false, a0, false, b2, (short)0, acc02, false, false);
        acc03 = __builtin_amdgcn_wmma_f32_16x16x4_f32(false, a0, false, b3, (short)0, acc03, false, false);
        acc10 = __builtin_amdgcn_wmma_f32_16x16x4_f32(false, a1, false, b0, (short)0, acc10, false, false);
        acc11 = __builtin_amdgcn_wmma_f32_16x16x4_f32(false, a1, false, b1, (short)0, acc11, false, false);
        acc12 = __builtin_amdgcn_wmma_f32_16x16x4_f32(false, a1, false, b2, (short)0, acc12, false, false);
        acc13 = __builtin_amdgcn_wmma_f32_16x16x4_f32(false, a1, false, b3, (short)0, acc13, false, false);
      }
    }
  }

  float* outB = out + (size_t)blockIdx.z * (size_t)outBatchStride;
  const float* biasB = bias ? bias + (size_t)blockIdx.z * (size_t)biasBatchStride : nullptr;

#pragma unroll
  for (int r = 0; r < 8; ++r) {
    // M-subtile 0
    {
      const int row = m0 + r + 8 * half;
      if (row < outC) {
        float badd = biasB ? biasB[row] : 0.f;
        float v0 = acc00[r] + badd;
        float v1 = acc01[r] + badd;
        float v2 = acc02[r] + badd;
        float v3 = acc03[r] + badd;
        if (relu) {
          v0 = fmaxf(v0, 0.f); v1 = fmaxf(v1, 0.f);
          v2 = fmaxf(v2, 0.f); v3 = fmaxf(v3, 0.f);
        }
        outB[(size_t)row * N + n0 +  0 + lid] = v0;
        outB[(size_t)row * N + n0 + 16 + lid] = v1;
        outB[(size_t)row * N + n0 + 32 + lid] = v2;
        outB[(size_t)row * N + n0 + 48 + lid] = v3;
      }
    }
    // M-subtile 1
    {
      const int row = m0 + 16 + r + 8 * half;
      if (row < outC) {
        float badd = biasB ? biasB[row] : 0.f;
        float v0 = acc10[r] + badd;
        float v1 = acc11[r] + badd;
        float v2 = acc12[r] + badd;
        float v3 = acc13[r] + badd;
        if (relu) {
          v0 = fmaxf(v0, 0.f); v1 = fmaxf(v1, 0.f);
          v2 = fmaxf(v2, 0.f); v3 = fmaxf(v3, 0.f);
        }
        outB[(size_t)row * N + n0 +  0 + lid] = v0;
        outB[(size_t)row * N + n0 + 16 + lid] = v1;
        outB[(size_t)row * N + n0 + 32 + lid] = v2;
        outB[(size_t)row * N + n0 + 48 + lid] = v3;
      }
    }
  }
}

// ---------------------------------------------------------------------------
// spatial mean: v[b*DIMS+d] = mean over `spatial` of h[(b*DIMS+d)*spatial + :]
// ---------------------------------------------------------------------------
__global__ void mean_kernel(const float* __restrict__ h,
                            float* __restrict__ v, int spatial) {
  int idx = blockIdx.x * blockDim.x + threadIdx.x;
  if (idx >= BB*DIMS) return;
  const float* p = h + (size_t)idx * spatial;
  float s = 0.f;
  for (int i = 0; i < spatial; ++i) s += p[i];
  v[idx] = s / (float)spatial;
}

// ---------------------------------------------------------------------------
// tiny FC (M=4 rows): out[b,m] = sum_d v[b,d]*W[m,d] + bias[m]
// ---------------------------------------------------------------------------
__global__ void fc_kernel(const float* __restrict__ v,
                          const float* __restrict__ W,
                          const float* __restrict__ bias,
                          float* __restrict__ out, int Mrows, int D) {
  int idx = blockIdx.x * blockDim.x + threadIdx.x;
  if (idx >= BB * Mrows) return;
  int b = idx / Mrows;
  int m = idx - b * Mrows;
  const float* vb = v + (size_t)b * D;
  const float* wr = W + (size_t)m * D;
  float s = bias[m];
  for (int d = 0; d < D; ++d) s += vb[d] * wr[d];
  out[idx] = s;
}

// ---------------------------------------------------------------------------
// deformable bilinear sampling for one batch:
//   samp[c*9+t, p] = bilinear(xe_b[c], h + t/3-1 + off[t,0,p], w + t%3-1 + off[t,1,p])
// zero outside the image (matches reference valid-mask semantics)
// ---------------------------------------------------------------------------
__global__ void deform_sample_kernel(const float* __restrict__ xe_b,
                                     const float* __restrict__ off_b,
                                     float* __restrict__ samp) {
  int idx = blockIdx.x * blockDim.x + threadIdx.x;
  if (idx >= DIMS*9*HW) return;
  int p = idx & (HW - 1);
  int t = (idx >> 12) % 9;
  int c = idx / (9*HW);
  int w = p & 63, h = p >> 6;
  float oy = off_b[(size_t)(t*2 + 0)*HW + p];
  float ox = off_b[(size_t)(t*2 + 1)*HW + p];
  float ys = (float)h + (float)(t/3 - 1) + oy;
  float xs = (float)w + (float)(t%3 - 1) + ox;
  float y0f = floorf(ys), x0f = floorf(xs);
  float wy = ys - y0f, wx = xs - x0f;
  int y0 = (int)y0f, x0 = (int)x0f;
  const float* ch = xe_b + (size_t)c*HW;
  auto g = [&](int yi, int xi) -> float {
    if (yi < 0 || yi >= HH || xi < 0 || xi >= WW) return 0.f;
    return ch[yi*WW + xi];
  };
  samp[idx] = g(y0,   x0  ) * (1.f - wy) * (1.f - wx)
            + g(y0,   x0+1) * (1.f - wy) * wx
            + g(y0+1, x0  ) * wy * (1.f - wx)
            + g(y0+1, x0+1) * wy * wx;
}

// ---------------------------------------------------------------------------
// host driver
// ---------------------------------------------------------------------------
static inline int blocks_for(int n) { return (n + 255) / 256; }

extern "C" void kernel_launch(void* const* d_in, const int* in_sizes, int n_in,
                              void* d_out, int out_size, void* d_ws, size_t ws_size,
                              hipStream_t stream) {
  (void)in_sizes; (void)n_in; (void)out_size; (void)ws_size;

  const float* R0     = (const float*)d_in[0];
  const float* T0     = (const float*)d_in[1];
  const float* inputs = (const float*)d_in[2];
  const float* po_w[2]  = { (const float*)d_in[3],  (const float*)d_in[4]  };
  const float* c1w[2]   = { (const float*)d_in[5],  (const float*)d_in[12] };
  const float* c2w[2]   = { (const float*)d_in[6],  (const float*)d_in[13] };
  const float* c3w[2]   = { (const float*)d_in[7],  (const float*)d_in[14] };
  const float* wxw[2]   = { (const float*)d_in[8],  (const float*)d_in[15] };
  const float* wxb[2]   = { (const float*)d_in[9],  (const float*)d_in[16] };
  const float* bxw[2]   = { (const float*)d_in[10], (const float*)d_in[17] };
  const float* bxb[2]   = { (const float*)d_in[11], (const float*)d_in[18] };
  const float* encw[2]  = { (const float*)d_in[19], (const float*)d_in[21] };
  const float* encb[2]  = { (const float*)d_in[20], (const float*)d_in[22] };
  const float* fusion_w = (const float*)d_in[23];

  // workspace carve (floats); total ~35.2M floats ~= 141 MB
  float* ws = (float*)d_ws;
  float* Rp     = ws; ws += (size_t)BB*CC*HW;          // 4,194,304
  float* Rc     = ws; ws += (size_t)BB*CC*HW;
  float* Tc     = ws; ws += (size_t)BB*CC*HW;
  float* corr   = ws; ws += (size_t)BB*CORR*HW;        //   802,816 (per stream, reused)
  float* ifw    = ws; ws += (size_t)BB*IFWC*HW;        // 9,191,424 (reused)
  float* offb   = ws; ws += (size_t)BB*18*HW;          //   294,912 (reused)
  float* h1     = ws; ws += (size_t)BB*DIMS*32*32;     //   524,288
  float* h2     = ws; ws += (size_t)BB*DIMS*16*16;     //   131,072
  float* h3     = ws; ws += (size_t)BB*DIMS*8*8;       //    32,768
  float* vvec   = ws; ws += (size_t)BB*DIMS;           //       512
  float* wmat   = ws; ws += (size_t)BB*WROW;           //   589,824
  float* bvec   = ws; ws += (size_t)BB*DIMS;           //       512
  float* xe     = ws; ws += (size_t)BB*DIMS*HW;        // 2,097,152
  float* samp   = ws; ws += (size_t)DKK*HW;            // 4,718,592 (per batch, reused)
  float* outcat = ws; ws += (size_t)BB*2*DIMS*HW;      // 4,194,304

  const int T = 256;

  // pooled features (persist across both streams)
  pool_kernel<<<blocks_for(BB*CC*HW), T, 0, stream>>>(R0, Rp, 0);
  pool_kernel<<<blocks_for(BB*CC*HW), T, 0, stream>>>(R0, Rc, 1);
  pool_kernel<<<blocks_for(BB*CC*HW), T, 0, stream>>>(T0, Tc, 1);

  for (int s = 0; s < 2; ++s) {
    const float* Acorr = s ? Rc : Rp;

    corr_kernel<<<blocks_for(BB*CORR*HW), T, 0, stream>>>(Acorr, Tc, corr);
    concat_kernel<<<blocks_for(BB*IFWC*HW), T, 0, stream>>>(corr, inputs, s, ifw);

    // offset conv: 305ch (first part of ifw), 3x3 s1 p1 -> 18ch 64x64
    {
      dim3 g(1, HW/64, BB);
      conv_gemm_wmma<<<g, 32, 0, stream>>>(ifw, IFWC*HW, po_w[s], 0, nullptr, 0,
          offb, 18*HW, IFOC, HH, WW, 18, HH, WW, 3, 3, 1, 1, 0);
    }
    // predictor stack
    {
      dim3 g(DIMS/32, (32*32)/64, BB);
      conv_gemm_wmma<<<g, 32, 0, stream>>>(ifw, IFWC*HW, c1w[s], 0, nullptr, 0,
          h1, DIMS*32*32, IFWC, HH, WW, DIMS, 32, 32, 5, 5, 2, 2, 1);
    }
    {
      dim3 g(DIMS/32, (16*16)/64, BB);
      conv_gemm_wmma<<<g, 32, 0, stream>>>(h1, DIMS*32*32, c2w[s], 0, nullptr, 0,
          h2, DIMS*16*16, DIMS, 32, 32, DIMS, 16, 16, 5, 5, 2, 2, 1);
    }
    {
      dim3 g(DIMS/32, (8*8)/64, BB);
      conv_gemm_wmma<<<g, 32, 0, stream>>>(h2, DIMS*16*16, c3w[s], 0, nullptr, 0,
          h3, DIMS*8*8, DIMS, 16, 16, DIMS, 8, 8, 3, 3, 2, 1, 0);
    }
    mean_kernel<<<blocks_for(BB*DIMS), T, 0, stream>>>(h3, vvec, 64);

    fc_kernel<<<blocks_for(BB*WROW), T, 0, stream>>>(vvec, wxw[s], wxb[s], wmat, WROW, DIMS);
    fc_kernel<<<blocks_for(BB*DIMS), T, 0, stream>>>(vvec, bxw[s], bxb[s], bvec, DIMS, DIMS);

    // encoder 1x1 (+bias): x for s=0, y for s=1
    {
      dim3 g(DIMS/32, HW/64, BB);
      conv_gemm_wmma<<<g, 32, 0, stream>>>(inputs + (size_t)s*CC*HW, 2*CC*HW,
          encw[s], 0, encb[s], 0,
          xe, DIMS*HW, CC, HH, WW, DIMS, HH, WW, 1, 1, 1, 0, 0);
    }

    // deformable conv: sample per batch, then predicted-weight GEMM into outcat
    for (int b = 0; b < BB; ++b) {
      deform_sample_kernel<<<blocks_for(DIMS*9*HW), T, 0, stream>>>(
          xe + (size_t)b*DIMS*HW, offb + (size_t)b*18*HW, samp);
      dim3 g(DIMS/32, HW/64, 1);
      conv_gemm_wmma<<<g, 32, 0, stream>>>(samp, 0,
          wmat + (size_t)b*WROW, 0,
          bvec + (size_t)b*DIMS, 0,
          outcat + (size_t)b*2*DIMS*HW + (size_t)s*DIMS*HW, 0,
          DKK, HH, WW, DIMS, HH, WW, 1, 1, 1, 0, 0);
    }
  }

  // fusion conv: (B, 256, 64, 64) -> (B, 256, 64, 64), 3x3 p1
  {
    dim3 g(CC/32, HW/64, BB);
    conv_gemm_wmma<<<g, 32, 0, stream>>>(outcat, 2*DIMS*HW, fusion_w, 0, nullptr, 0,
        (float*)d_out, CC*HW, 2*DIMS, HH, WW, CC, HH, WW, 3, 3, 1, 1, 0);
  }
}